// GINWithoutNoise_69441031242295
// MI455X (gfx1250) — compile-verified
//
#include <hip/hip_runtime.h>
#include <hip/hip_bf16.h>

typedef float v2f __attribute__((ext_vector_type(2)));
typedef float v8f __attribute__((ext_vector_type(8)));

#define K_DIM 128

// ---- async global->LDS staging helpers (CDNA5 GLOBAL_LOAD_ASYNC_TO_LDS_B128) ----
#if defined(__AMDGCN__) && __has_builtin(__builtin_amdgcn_global_load_async_to_lds_b128)
#define GIN_ASYNC_LDS 1
#else
#define GIN_ASYNC_LDS 0
#endif

#if GIN_ASYNC_LDS
typedef int gin_int4 __attribute__((vector_size(4 * sizeof(int))));
#endif

__device__ __forceinline__ void stage16(float* ldst, const float* gsrc) {
#if GIN_ASYNC_LDS
  __builtin_amdgcn_global_load_async_to_lds_b128(
      (__attribute__((address_space(1))) gin_int4*)gsrc,
      (__attribute__((address_space(3))) gin_int4*)ldst,
      /*imm offset=*/0, /*cpol=*/0);
#else
  *(float4*)ldst = *(const float4*)gsrc;
#endif
}

__device__ __forceinline__ void stage_join() {
#if GIN_ASYNC_LDS
#if __has_builtin(__builtin_amdgcn_s_wait_asynccnt)
  __builtin_amdgcn_s_wait_asynccnt(0);
#else
  asm volatile("s_wait_asynccnt 0x0" ::: "memory");
#endif
#endif
}

// t0[s,n,k] = (1+eps0) * (k<64 ? X[n,k] : h[s,n,k-64])
// (fused concat + eps-scale + aggregation-buffer init)
__global__ __launch_bounds__(256) void gin_build_t0(
    const float* __restrict__ X, const float* __restrict__ h,
    const float* __restrict__ eps0p, float* __restrict__ t0,
    int N, int totalV4) {
  int idx = blockIdx.x * 256 + threadIdx.x;
  if (idx >= totalV4) return;
  float scale = 1.0f + eps0p[0];
  int k4 = idx & 31;        // which float4 within the 128-dim row
  long sn = idx >> 5;       // s*N + n
  int k = k4 << 2;
  float4 v;
  if (k < 64) {
    long n = sn % N;
    v = *(const float4*)(X + n * 64 + k);
  } else {
    v = *(const float4*)(h + sn * 64 + (k - 64));   // h is [S,N,64] => flat sn*64
  }
  float4 o; o.x = v.x*scale; o.y = v.y*scale; o.z = v.z*scale; o.w = v.w*scale;
  *(float4*)(t0 + (long)idx * 4) = o;
}

// One wave per edge; lane L owns feature dims [4L,4L+4). Gathers straight from X/h.
__global__ __launch_bounds__(256) void gin_scatter0(
    const float* __restrict__ X, const float* __restrict__ h,
    const int* __restrict__ src, const int* __restrict__ dst,
    float* __restrict__ t0, int N, int E, int S) {
  int gt = blockIdx.x * 256 + threadIdx.x;
  int e = gt >> 5, lane = gt & 31;
  if (e >= E) return;
  int u = src[e], v = dst[e];
  int d0 = lane << 2;
  for (int s = 0; s < S; ++s) {
    float4 val;
    if (d0 < 64) val = *(const float4*)(X + (long)u * 64 + d0);
    else         val = *(const float4*)(h + ((long)s * N + u) * 64 + (d0 - 64));
    float* tp = t0 + ((long)s * N + v) * K_DIM + d0;
    atomicAdd(tp + 0, val.x);
    atomicAdd(tp + 1, val.y);
    atomicAdd(tp + 2, val.z);
    atomicAdd(tp + 3, val.w);
  }
}

// Layer-1 scatter: gather from x1 [S,N,128], accumulate into t1.
__global__ __launch_bounds__(256) void gin_scatter1(
    const float* __restrict__ x1,
    const int* __restrict__ src, const int* __restrict__ dst,
    float* __restrict__ t1, int N, int E, int S) {
  int gt = blockIdx.x * 256 + threadIdx.x;
  int e = gt >> 5, lane = gt & 31;
  if (e >= E) return;
  int u = src[e], v = dst[e];
  int d0 = lane << 2;
  for (int s = 0; s < S; ++s) {
    float4 val = *(const float4*)(x1 + ((long)s * N + u) * K_DIM + d0);
    float* tp = t1 + ((long)s * N + v) * K_DIM + d0;
    atomicAdd(tp + 0, val.x);
    atomicAdd(tp + 1, val.y);
    atomicAdd(tp + 2, val.z);
    atomicAdd(tp + 3, val.w);
  }
}

// Y[M,F] = A[M,128] @ W[128,F] + bias; if HAS_T: T[M,128] = (1+eps1)*Y (T may alias A).
// 256 threads = 8 waves; each wave owns TWO 16x16 output tiles for operand reuse:
//   F=128: 2 N-tiles per wave (reuse A fragment), block = 32 rows x 128 cols
//   F=64 : 2 M-tiles per wave (reuse B fragment), block = 64 rows x  64 cols
// LDS: W (K*F) + block A-tile, stride 132 floats (16B-aligned chunks, bank-conflict free).
template <int F, bool HAS_T>
__global__ __launch_bounds__(256) void gin_gemm_wmma(
    const float* __restrict__ A, const float* __restrict__ Wg,
    const float* __restrict__ bias, float* __restrict__ Y,
    float* __restrict__ T, const float* __restrict__ eps1p, long M) {
  extern __shared__ float lds[];
  constexpr int K   = K_DIM;
  constexpr int LDA = K + 4;                  // 132: keeps 16B alignment + no bank conflicts
  constexpr int NT  = F / 16;                 // n-tiles total: 8 or 4
  constexpr int NPW = (F == 128) ? 2 : 1;     // n-tiles per wave
  constexpr int MPW = (F == 128) ? 1 : 2;     // m-tiles per wave
  constexpr int NG  = NT / NPW;               // 4 n-groups of waves
  constexpr int MG  = 8 / NG;                 // 2 m-groups of waves
  constexpr int ROWS = MG * MPW * 16;         // 32 or 64 rows per block

  float* lW = lds;                            // K*F floats
  float* lA = lds + K * F;                    // ROWS * LDA floats

  int tid = threadIdx.x;
  long row0 = (long)blockIdx.x * ROWS;
  bool full = (row0 + ROWS) <= M;             // block-uniform

  // stage W (async b128 chunks)
  constexpr int WCH = (K * F) / 4;
  #pragma unroll
  for (int c = tid; c < WCH; c += 256) stage16(lW + c * 4, Wg + c * 4);
  // stage A tile
  constexpr int ACH = (ROWS * K) / 4;
  for (int c = tid; c < ACH; c += 256) {
    int r = c >> 5, kq = c & 31;              // K/4 == 32 chunks per row
    long rr = row0 + r;
    if (rr >= M) rr = M - 1;
    stage16(lA + r * LDA + kq * 4, A + rr * K + kq * 4);
  }
  stage_join();       // s_wait_asynccnt 0 -> all global reads of A done (in-place T safe)
  __syncthreads();

  int wid = tid >> 5, lane = tid & 31;
  int half = lane >> 4, col = lane & 15;
  int mg = wid / NG, ng = wid - mg * NG;

  v8f acc[MPW * NPW];
  #pragma unroll
  for (int i = 0; i < MPW * NPW; ++i) acc[i] = v8f{0.f,0.f,0.f,0.f,0.f,0.f,0.f,0.f};

  const float* aRow[MPW];
  #pragma unroll
  for (int mi = 0; mi < MPW; ++mi)
    aRow[mi] = lA + ((mg * MPW + mi) * 16 + col) * LDA;

  int cf[NPW];
  #pragma unroll
  for (int ni = 0; ni < NPW; ++ni) cf[ni] = (ng + ni * NG) * 16 + col;

  // A 16x4 f32 fragment: lane(col) = row; half 0 -> K={k0,k0+1}, half 1 -> K={k0+2,k0+3}
  // B 4x16 f32 fragment mirrored: b.x = W[kk][cf], b.y = W[kk+1][cf]
  #pragma unroll 4
  for (int k0 = 0; k0 < K; k0 += 4) {
    int kk = k0 + (half << 1);
    v2f a[MPW], b[NPW];
    #pragma unroll
    for (int mi = 0; mi < MPW; ++mi) a[mi] = *(const v2f*)(aRow[mi] + kk);
    #pragma unroll
    for (int ni = 0; ni < NPW; ++ni) {
      b[ni].x = lW[kk * F + cf[ni]];
      b[ni].y = lW[(kk + 1) * F + cf[ni]];
    }
    #pragma unroll
    for (int mi = 0; mi < MPW; ++mi)
      #pragma unroll
      for (int ni = 0; ni < NPW; ++ni)
        acc[mi * NPW + ni] = __builtin_amdgcn_wmma_f32_16x16x4_f32(
            false, a[mi], false, b[ni], (short)0, acc[mi * NPW + ni], false, false);
  }

  float e1 = 0.f;
  if constexpr (HAS_T) e1 = 1.0f + eps1p[0];

  // C/D layout: VGPR j -> row (j + 8*half), col = cf
  #pragma unroll
  for (int mi = 0; mi < MPW; ++mi) {
    long rbase = row0 + (mg * MPW + mi) * 16 + (half << 3);
    #pragma unroll
    for (int ni = 0; ni < NPW; ++ni) {
      float bv = bias[cf[ni]];
      const v8f& c = acc[mi * NPW + ni];
      if (full) {
        #pragma unroll
        for (int j = 0; j < 8; ++j) {
          long r = rbase + j;
          float y = c[j] + bv;
          Y[r * F + cf[ni]] = y;
          if constexpr (HAS_T) T[r * K + cf[ni]] = e1 * y;
        }
      } else {
        #pragma unroll
        for (int j = 0; j < 8; ++j) {
          long r = rbase + j;
          if (r < M) {
            float y = c[j] + bv;
            Y[r * F + cf[ni]] = y;
            if constexpr (HAS_T) T[r * K + cf[ni]] = e1 * y;
          }
        }
      }
    }
  }
}

extern "C" void kernel_launch(void* const* d_in, const int* in_sizes, int n_in,
                              void* d_out, int out_size, void* d_ws, size_t ws_size,
                              hipStream_t stream) {
  const float* X   = (const float*)d_in[0];
  const float* h   = (const float*)d_in[1];
  const float* W0  = (const float*)d_in[2];
  const float* b0  = (const float*)d_in[3];
  const float* e0  = (const float*)d_in[4];
  const float* W1  = (const float*)d_in[5];
  const float* b1  = (const float*)d_in[6];
  const float* e1  = (const float*)d_in[7];
  const int*   src = (const int*)d_in[8];
  const int*   dst = (const int*)d_in[9];

  int N = in_sizes[0] / 64;
  int S = in_sizes[1] / in_sizes[0];
  int E = in_sizes[8];
  long M = (long)S * N;
  int F1 = out_size / (int)M;            // 64

  float* t0 = (float*)d_ws;              // [M,128], becomes t1 in place after gemm0
  float* x1 = t0 + M * K_DIM;            // [M,128]

  int totalV4 = (int)(M * (K_DIM / 4));
  gin_build_t0<<<(totalV4 + 255) / 256, 256, 0, stream>>>(X, h, e0, t0, N, totalV4);

  long sthreads = (long)E * 32;
  int sblocks = (int)((sthreads + 255) / 256);
  gin_scatter0<<<sblocks, 256, 0, stream>>>(X, h, src, dst, t0, N, E, S);

  size_t smem0 = (size_t)(K_DIM * K_DIM + 32 * (K_DIM + 4)) * sizeof(float);   // ~80.5 KB
  int g0 = (int)((M + 31) / 32);
  gin_gemm_wmma<128, true><<<g0, 256, smem0, stream>>>(t0, W0, b0, x1, t0, e1, M);

  gin_scatter1<<<sblocks, 256, 0, stream>>>(x1, src, dst, t0, N, E, S);

  size_t smem1 = (size_t)(K_DIM * 64 + 64 * (K_DIM + 4)) * sizeof(float);      // ~65 KB
  int g1 = (int)((M + 63) / 64);
  gin_gemm_wmma<64, false><<<g1, 256, smem1, stream>>>(t0, W1, b1, (float*)d_out,
                                                       nullptr, nullptr, M);
}